// SpectralAttention_44659069944188
// MI455X (gfx1250) — compile-verified
//
#include <hip/hip_runtime.h>
#include <math.h>

typedef float v2f __attribute__((ext_vector_type(2)));
typedef float v8f __attribute__((ext_vector_type(8)));

#define N_LEN  4096          // N
#define NROWS  8192          // B*C
#define NC     4094          // valid centers (N-2)
#define KDIM   8192          // padded concat counts: peak[0..4095] | valley[4096..8191]
#define KW     16            // K-parallel waves per output tile
#define KCHUNK (KDIM / KW)   // 512 (aligned: never straddles the 4096 boundary)

// ---------------------------------------------------------------------------
// Kernel 0: zero the count accumulators (ws is poisoned once, never re-zeroed)
// ---------------------------------------------------------------------------
__global__ void __launch_bounds__(256)
zero_counts_kernel(unsigned* __restrict__ counts) {
    int i = blockIdx.x * 256 + threadIdx.x;
    if (i < KDIM) counts[i] = 0u;
}

// ---------------------------------------------------------------------------
// Kernel 1: peak/valley counts. Thread owns 4 consecutive columns; per row it
// issues one b128 + one b64 load (instead of 3 scalar loads per column).
// grid = (4 column blocks, 32 row groups of 256 rows), block = 256.
// Flush: <=8 global atomics per thread (~256K total).
// ---------------------------------------------------------------------------
__global__ void __launch_bounds__(256)
count_kernel(const float* __restrict__ x, unsigned* __restrict__ counts) {
    const int t  = blockIdx.x * 256 + threadIdx.x;   // 0..1023
    const int j4 = t * 4;                            // first column (0..4092)
    const int rows_per = NROWS / gridDim.y;          // 256
    const int r0 = blockIdx.y * rows_per;
    const float* xp = x + (size_t)r0 * N_LEN + j4;
    const bool has_tail = (j4 < N_LEN - 4);          // may we read x[j4+4..5]?

    unsigned pc0 = 0, pc1 = 0, pc2 = 0, pc3 = 0;
    unsigned vc0 = 0, vc1 = 0, vc2 = 0, vc3 = 0;
    for (int r = 0; r < rows_per; ++r) {
        float4 v = *(const float4*)xp;               // x[j4..j4+3], 16B aligned
        float2 w = {0.0f, 0.0f};
        if (has_tail) w = *(const float2*)(xp + 4);  // x[j4+4..j4+5], 8B aligned
        float s0 = (v.z - v.y) - (v.y - v.x);
        float s1 = (v.w - v.z) - (v.z - v.y);
        float s2 = (w.x - v.w) - (v.w - v.z);
        float s3 = (w.y - w.x) - (w.x - v.w);
        pc0 += (s0 < 0.0f); vc0 += (s0 > 0.0f);
        pc1 += (s1 < 0.0f); vc1 += (s1 > 0.0f);
        pc2 += (s2 < 0.0f); vc2 += (s2 > 0.0f);
        pc3 += (s3 < 0.0f); vc3 += (s3 > 0.0f);
        xp += N_LEN;
    }
    unsigned pcs[4] = {pc0, pc1, pc2, pc3};
    unsigned vcs[4] = {vc0, vc1, vc2, vc3};
    #pragma unroll
    for (int i = 0; i < 4; ++i) {
        int j = j4 + i;
        if (j < NC) {
            if (pcs[i]) atomicAdd(&counts[j], pcs[i]);
            if (vcs[i]) atomicAdd(&counts[4096 + j], vcs[i]);
        }
    }
}

// ---------------------------------------------------------------------------
// Kernel 2: vec = [peak;valley] (1x8192, M padded to 16) @ G (8192x4096)
// via V_WMMA_F32_16X16X4_F32. 16 waves per 16-wide output tile; each wave
// owns a 512-wide K chunk (entirely peak OR valley, so the width/center
// select hoists out of the loop) -> 128 WMMAs per wave, LDS reduction after.
// A fragment loaded as one unconditional ds_load_b64 then v_cndmask'ed
// (no exec-masked loads). Gaussian args updated incrementally (d -= 4).
// ---------------------------------------------------------------------------
__global__ void __launch_bounds__(KW * 32)
wmma_vec_kernel(const unsigned* __restrict__ counts,
                const float* __restrict__ log_pw,
                const float* __restrict__ log_vw,
                float* __restrict__ vec) {
    __shared__ float cnt[KDIM];                  // 32 KB
    __shared__ float part[KW * 16];              // 1 KB partial sums

    const int tid  = threadIdx.x;
    const int wave = tid >> 5;                   // 0..15
    const int lane = tid & 31;
    const int half = lane >> 4;                  // 0: K pair (0,1); 1: (2,3)
    const int col  = lane & 15;
    const int tile = blockIdx.x;                 // 0..255
    const int p    = tile * 16 + col;            // output position

    // Stage this wave's K chunk of counts into LDS (u32 -> f32).
    const int kchunk = wave * KCHUNK;
    for (int i = lane; i < KCHUNK; i += 32)
        cnt[kchunk + i] = (float)counts[kchunk + i];
    __syncthreads();

    // (d/w)^2 = d^2 * exp(-2*log_w); whole chunk is one gaussian family.
    const bool  valley = (kchunk >= 4096);
    const float m  = -__expf(-2.0f * (valley ? log_vw[0] : log_pw[0]));
    const int   off = valley ? 4096 : 0;         // k -> center j = k - off

    // d(k) = p - (k - off) - 1; per iteration k advances by 4.
    float d0 = (float)(p - 1 + off - kchunk - 2 * half);
    float d1 = d0 - 1.0f;
    const float2* cp = (const float2*)&cnt[kchunk + 2 * half];
    const bool rowA = (col == 0);                // only M=0 row of A nonzero

    v8f acc = {};
    for (int it = 0; it < KCHUNK / 4; ++it) {
        float2 c01 = cp[0];                      // one ds_load_b64, all lanes
        cp += 2;                                 // advance 4 floats
        v2f a;
        a.x = rowA ? c01.x : 0.0f;               // v_cndmask, no exec games
        a.y = rowA ? c01.y : 0.0f;
        v2f b;
        b.x = __expf(d0 * d0 * m);
        b.y = __expf(d1 * d1 * m);
        d0 -= 4.0f;
        d1 -= 4.0f;
        acc = __builtin_amdgcn_wmma_f32_16x16x4_f32(
            false, a, false, b, (short)0, acc, false, false);
    }

    if (half == 0) part[wave * 16 + col] = acc[0];   // M=0 row, lanes 0-15
    __syncthreads();
    if (tid < 16) {
        float s = 0.0f;
        #pragma unroll
        for (int w = 0; w < KW; ++w) s += part[w * 16 + tid];
        vec[tile * 16 + tid] = s;
    }
}

// ---------------------------------------------------------------------------
// Kernel 3: single block -> min/max of vec, write normalized vec.
// ---------------------------------------------------------------------------
__global__ void __launch_bounds__(256)
norm_kernel(const float* __restrict__ vec, float* __restrict__ norm) {
    __shared__ float smin[256];
    __shared__ float smax[256];
    const int t = threadIdx.x;
    float mn =  3.0e38f, mx = -3.0e38f;
    for (int i = t; i < N_LEN; i += 256) {
        float v = vec[i];
        mn = fminf(mn, v);
        mx = fmaxf(mx, v);
    }
    smin[t] = mn; smax[t] = mx;
    __syncthreads();
    for (int s = 128; s > 0; s >>= 1) {
        if (t < s) {
            smin[t] = fminf(smin[t], smin[t + s]);
            smax[t] = fmaxf(smax[t], smax[t + s]);
        }
        __syncthreads();
    }
    const float lo  = smin[0];
    const float den = smax[0] - lo + 1e-6f;
    for (int i = t; i < N_LEN; i += 256)
        norm[i] = (vec[i] - lo) / den;
}

// ---------------------------------------------------------------------------
// Kernel 4: broadcast normalized vec to all B*C rows (float4 stores; the
// 16 KB source stays L1/L2-resident, stores are the 128 MiB HBM cost).
// ---------------------------------------------------------------------------
__global__ void __launch_bounds__(256)
bcast_kernel(const float* __restrict__ norm, float4* __restrict__ out) {
    size_t i = (size_t)blockIdx.x * 256 + threadIdx.x;   // float4 index
    const float4* n4 = (const float4*)norm;
    out[i] = n4[i & 1023u];                              // 1024 float4 per row
}

// ---------------------------------------------------------------------------
extern "C" void kernel_launch(void* const* d_in, const int* in_sizes, int n_in,
                              void* d_out, int out_size, void* d_ws, size_t ws_size,
                              hipStream_t stream) {
    const float* x   = (const float*)d_in[0];
    const float* lpw = (const float*)d_in[1];
    const float* lvw = (const float*)d_in[2];
    float* out = (float*)d_out;

    char* ws = (char*)d_ws;
    unsigned* counts = (unsigned*)ws;                    // 8192 u32  = 32 KB
    float*    vec    = (float*)(ws + 32768);             // 4096 f32  = 16 KB
    float*    norm   = (float*)(ws + 49152);             // 4096 f32  = 16 KB

    zero_counts_kernel<<<KDIM / 256, 256, 0, stream>>>(counts);
    count_kernel<<<dim3(4, 32), 256, 0, stream>>>(x, counts);
    wmma_vec_kernel<<<N_LEN / 16, KW * 32, 0, stream>>>(counts, lpw, lvw, vec);
    norm_kernel<<<1, 256, 0, stream>>>(vec, norm);

    const size_t n4 = (size_t)out_size / 4;              // 8,388,608 float4
    bcast_kernel<<<(unsigned)(n4 / 256), 256, 0, stream>>>(norm, (float4*)out);
}